// Net_49632642073421
// MI455X (gfx1250) — compile-verified
//
#include <hip/hip_runtime.h>
#include <stdint.h>

#define NNODES 100000
#define NEDGES 3200000
#define XS 128  // padded feature stride

typedef float v2f __attribute__((ext_vector_type(2)));
typedef float v8f __attribute__((ext_vector_type(8)));
typedef unsigned int u32x4 __attribute__((ext_vector_type(4)));
typedef int i32x4 __attribute__((ext_vector_type(4)));
typedef int i32x8 __attribute__((ext_vector_type(8)));

#if !defined(__has_builtin)
#define __has_builtin(x) 0
#endif
#if __has_builtin(__builtin_amdgcn_tensor_load_to_lds) && __has_builtin(__builtin_amdgcn_s_wait_tensorcnt)
#define HAVE_TDM 1
#else
#define HAVE_TDM 0
#endif

// ---------------------------------------------------------------- init
__global__ void init_kernel(float* __restrict__ x, float* __restrict__ deg, int n) {
    int i = blockIdx.x * blockDim.x + threadIdx.x;
    int total = n * XS;
    if (i < total) x[i] = ((i & (XS - 1)) == 0) ? 1.0f : 0.0f;  // x0 = ones(N,1), pad 0
    if (i < n) deg[i] = 1.0f;                                   // self-loop
}

__global__ void degree_kernel(const int* __restrict__ dst, float* __restrict__ deg, int e) {
    int i = blockIdx.x * blockDim.x + threadIdx.x;
    if (i < e) atomicAdd(&deg[dst[i]], 1.0f);
}

__global__ void rsqrt_kernel(const float* __restrict__ deg, float* __restrict__ dis, int n) {
    int i = blockIdx.x * blockDim.x + threadIdx.x;
    if (i < n) dis[i] = rsqrtf(deg[i]);  // deg >= 1 always (self-loops)
}

// ---------------------------------------------------------------- TDM: DMA whole W [din x dout] into LDS
#if HAVE_TDM
__device__ __forceinline__ void tdm_load_W(const float* W, float* Wl, int din, int dout) {
    // D# group 0: count=1 | lds_addr | global_addr[56:0] | type=2
    unsigned lds = (unsigned)(uintptr_t)Wl;  // flat low 32 bits == wave-relative LDS byte addr
    unsigned long long ga = (unsigned long long)(uintptr_t)W;
    u32x4 g0;
    g0[0] = 1u;                                                // count=1, is_restore=0, gather off
    g0[1] = lds;                                               // lds_addr (bytes)
    g0[2] = (unsigned)ga;                                      // global_addr[31:0]
    g0[3] = (unsigned)((ga >> 32) & 0x01FFFFFFu) | (2u << 30); // global_addr[56:32], type=2
    // D# group 1: 2-D tensor, data_size=4B, tile == whole tensor
    unsigned d0 = (unsigned)dout, d1 = (unsigned)din;
    i32x8 g1;
    g1[0] = (int)(2u << 16);                 // workgroup_mask=0, data_size=2 (4 bytes)
    g1[1] = (int)(d0 << 16);                 // atomic_barrier_addr=0, tensor_dim0[15:0]
    g1[2] = (int)((d0 >> 16) | (d1 << 16));  // tensor_dim0[31:16] | tensor_dim1[15:0]
    g1[3] = (int)((d1 >> 16) | (d0 << 16));  // tensor_dim1[31:16] | tile_dim0=dout
    g1[4] = (int)d1;                         // tile_dim1=din, tile_dim2=0
    g1[5] = (int)d0;                         // tensor_dim0_stride[31:0]=dout
    g1[6] = 0;                               // stride hi | tensor_dim1_stride lo (unused)
    g1[7] = 0;
    i32x4 z4 = {0, 0, 0, 0};                 // groups 2/3: 2-D tile, all zero
#if defined(__clang_major__) && __clang_major__ >= 23
    i32x8 z8 = {0, 0, 0, 0, 0, 0, 0, 0};
    __builtin_amdgcn_tensor_load_to_lds(g0, g1, z4, z4, z8, 0);
#else
    __builtin_amdgcn_tensor_load_to_lds(g0, g1, z4, z4, 0);
#endif
    __builtin_amdgcn_s_wait_tensorcnt(0);
}
#endif

// ---------------------------------------------------------------- GEMM: H = X @ W  (fp32 WMMA)
// X: [n, XS] padded (cols >= din are zero).  W: [din, dout] row-major, DMA'd whole into LDS.
// H: [n, XS], cols [0, NTILES*16) written (cols >= dout are don't-care, never read).
template <int NTILES>
__global__ void __launch_bounds__(256)
gemm_wmma_kernel(const float* __restrict__ X, const float* __restrict__ W,
                 float* __restrict__ H, int din, int dout, int n) {
    __shared__ float Wl[16128];  // max Kpad*dout + 128 overrun window = 128*125+128

    const int tid   = threadIdx.x;
    const int wave  = tid >> 5;
    const int lane  = tid & 31;
    const int lhalf = lane & 15;
    const bool hi   = lane >= 16;  // lanes 16-31 carry K+2,K+3 of A / B fragments
    const int m0    = blockIdx.x * 128 + wave * 16;
    const int Kpad  = (din + 3) & ~3;

    // Zero the LDS tail (k in [din,Kpad) rows + edge-tile overrun) so padded-K WMMAs
    // see exact zeros (avoid 0 x NaN from uninitialized LDS). Disjoint from DMA region.
    {
        int zbeg = din * dout;
        int zend = Kpad * dout + 128;
        if (zend > 16128) zend = 16128;
        for (int idx = zbeg + tid; idx < zend; idx += 256) Wl[idx] = 0.0f;
    }
#if HAVE_TDM
    if (wave == 0) tdm_load_W(W, Wl, din, dout);  // one DMA per block; wave0 waits TENSORcnt
#else
    for (int idx = tid; idx < din * dout; idx += 256) Wl[idx] = W[idx];
#endif
    __syncthreads();

    // A fragment rows: clamp OOB rows to n-1 (their D rows are discarded at store time),
    // so the A load is one unconditional global_load_b64 per K-step.
    int row = m0 + lhalf;
    if (row >= n) row = n - 1;
    const float* xrow = X + (size_t)row * XS + (hi ? 2 : 0);

    v8f acc[NTILES];
#pragma unroll
    for (int t = 0; t < NTILES; ++t)
#pragma unroll
        for (int r = 0; r < 8; ++r) acc[t][r] = 0.0f;

    for (int kk = 0; kk < Kpad; kk += 4) {
        v2f a = *(const v2f*)(xrow + kk);  // lane<16 -> K,K+1 ; lane>=16 -> K+2,K+3
        const float* w0 = Wl + (kk + (hi ? 2 : 0)) * dout + lhalf;
#pragma unroll
        for (int t = 0; t < NTILES; ++t) {
            v2f b;
            b.x = w0[t * 16];         // B[kb  ][col]
            b.y = w0[t * 16 + dout];  // B[kb+1][col]
            acc[t] = __builtin_amdgcn_wmma_f32_16x16x4_f32(
                /*neg_a=*/false, a, /*neg_b=*/false, b,
                /*c_mod=*/(short)0, acc[t], /*reuse_a=*/false, /*reuse_b=*/false);
        }
    }

    // D layout: VGPR r -> M=r (lanes 0-15) / M=r+8 (lanes 16-31); N = lane&15 per tile
    if (m0 + 16 <= n) {  // wave-uniform fast path: no per-element guards
#pragma unroll
        for (int t = 0; t < NTILES; ++t) {
            float* hp = H + (size_t)(m0 + (hi ? 8 : 0)) * XS + t * 16 + lhalf;
#pragma unroll
            for (int r = 0; r < 8; ++r) hp[(size_t)r * XS] = acc[t][r];
        }
    } else {
#pragma unroll
        for (int t = 0; t < NTILES; ++t) {
            const int coln = t * 16 + lhalf;
#pragma unroll
            for (int r = 0; r < 8; ++r) {
                const int mrow = m0 + r + (hi ? 8 : 0);
                if (mrow < n) H[(size_t)mrow * XS + coln] = acc[t][r];
            }
        }
    }
}

// ---------------------------------------------------------------- y = h * dis^2 (self loop) + b, pad 0
__global__ void self_bias_kernel(const float* __restrict__ H, float* __restrict__ Y,
                                 const float* __restrict__ dis, const float* __restrict__ b,
                                 int dout, int n) {
    int i = blockIdx.x * blockDim.x + threadIdx.x;
    int total = n * XS;
    if (i >= total) return;
    int node = i >> 7;
    int j    = i & (XS - 1);
    float v = 0.0f;
    if (j < dout) {
        float d = dis[node];
        v = H[i] * d * d + b[j];
    }
    Y[i] = v;
}

// ---------------------------------------------------------------- edge scatter: one wave per edge
__global__ void scatter_kernel(const int* __restrict__ src, const int* __restrict__ dst,
                               const float* __restrict__ H, float* __restrict__ Y,
                               const float* __restrict__ dis, int dout, int e) {
    int wave = (blockIdx.x * blockDim.x + threadIdx.x) >> 5;
    int lane = threadIdx.x & 31;
    if (wave >= e) return;
    int s = src[wave];
    int d = dst[wave];
    float nrm = dis[s] * dis[d];
    const float* hrow = H + (size_t)s * XS;
    float* yrow = Y + (size_t)d * XS;
    for (int j = lane; j < dout; j += 32)
        atomicAdd(&yrow[j], hrow[j] * nrm);
}

// ---------------------------------------------------------------- min/max reduce + normalize
__global__ void minmax_partial_kernel(const float* __restrict__ X, float* __restrict__ pmin,
                                      float* __restrict__ pmax, int n) {
    __shared__ float smin[256], smax[256];
    float mn = 3.402823e38f, mx = -3.402823e38f;
    for (int i = blockIdx.x * blockDim.x + threadIdx.x; i < n; i += gridDim.x * blockDim.x) {
        float v = X[(size_t)i * XS];  // final dout == 1, value in col 0
        mn = fminf(mn, v);
        mx = fmaxf(mx, v);
    }
    int t = threadIdx.x;
    smin[t] = mn; smax[t] = mx;
    __syncthreads();
    for (int s = 128; s > 0; s >>= 1) {
        if (t < s) {
            smin[t] = fminf(smin[t], smin[t + s]);
            smax[t] = fmaxf(smax[t], smax[t + s]);
        }
        __syncthreads();
    }
    if (t == 0) { pmin[blockIdx.x] = smin[0]; pmax[blockIdx.x] = smax[0]; }
}

__global__ void minmax_final_kernel(const float* __restrict__ pmin, const float* __restrict__ pmax,
                                    float* __restrict__ mnmx) {
    __shared__ float smin[256], smax[256];
    int t = threadIdx.x;
    smin[t] = pmin[t]; smax[t] = pmax[t];
    __syncthreads();
    for (int s = 128; s > 0; s >>= 1) {
        if (t < s) {
            smin[t] = fminf(smin[t], smin[t + s]);
            smax[t] = fmaxf(smax[t], smax[t + s]);
        }
        __syncthreads();
    }
    if (t == 0) { mnmx[0] = smin[0]; mnmx[1] = smax[0]; }
}

__global__ void normalize_kernel(const float* __restrict__ X, const float* __restrict__ mnmx,
                                 float* __restrict__ out, int n) {
    int i = blockIdx.x * blockDim.x + threadIdx.x;
    if (i < n) out[i] = (X[(size_t)i * XS] - mnmx[0]) / (mnmx[1] - mnmx[0] + 1e-15f);
}

// ---------------------------------------------------------------- launcher
extern "C" void kernel_launch(void* const* d_in, const int* in_sizes, int n_in,
                              void* d_out, int out_size, void* d_ws, size_t ws_size,
                              hipStream_t stream) {
    const int N = NNODES, E = NEDGES;
    const int* edge = (const int*)d_in[0];
    const int* src = edge;
    const int* dst = edge + E;

    const float* W[6];
    const float* B[6];
    for (int l = 0; l < 6; ++l) {
        W[l] = (const float*)d_in[1 + 2 * l];
        B[l] = (const float*)d_in[2 + 2 * l];
    }
    static const int DIN[6]  = {1, 25, 125, 125, 125, 125};
    static const int DOUT[6] = {25, 125, 125, 125, 125, 1};
    static const int NT[6]   = {2, 8, 8, 8, 8, 1};  // ceil(dout/16)

    float* bufA = (float*)d_ws;           // current features x  (N x 128)
    float* bufB = bufA + (size_t)N * XS;  // GEMM output h       (N x 128)
    float* deg  = bufB + (size_t)N * XS;
    float* dis  = deg + N;
    float* pmin = dis + N;
    float* pmax = pmin + 256;
    float* mnmx = pmax + 256;
    float* out  = (float*)d_out;

    const int threads = 256;
    init_kernel<<<(N * XS + threads - 1) / threads, threads, 0, stream>>>(bufA, deg, N);
    degree_kernel<<<(E + threads - 1) / threads, threads, 0, stream>>>(dst, deg, E);
    rsqrt_kernel<<<(N + threads - 1) / threads, threads, 0, stream>>>(deg, dis, N);

    const int gemmBlocks = (N + 127) / 128;
    const int sbBlocks   = (N * XS + threads - 1) / threads;
    const int scBlocks   = (E + 7) / 8;  // 8 waves (edges) per 256-thread block

    for (int l = 0; l < 6; ++l) {
        switch (NT[l]) {
            case 1:
                gemm_wmma_kernel<1><<<gemmBlocks, threads, 0, stream>>>(bufA, W[l], bufB, DIN[l], DOUT[l], N);
                break;
            case 2:
                gemm_wmma_kernel<2><<<gemmBlocks, threads, 0, stream>>>(bufA, W[l], bufB, DIN[l], DOUT[l], N);
                break;
            default:
                gemm_wmma_kernel<8><<<gemmBlocks, threads, 0, stream>>>(bufA, W[l], bufB, DIN[l], DOUT[l], N);
                break;
        }
        self_bias_kernel<<<sbBlocks, threads, 0, stream>>>(bufB, bufA, dis, B[l], DOUT[l], N);
        scatter_kernel<<<scBlocks, threads, 0, stream>>>(src, dst, bufB, bufA, dis, DOUT[l], E);
    }

    minmax_partial_kernel<<<256, threads, 0, stream>>>(bufA, pmin, pmax, N);
    minmax_final_kernel<<<1, threads, 0, stream>>>(pmin, pmax, mnmx);
    normalize_kernel<<<(N + threads - 1) / threads, threads, 0, stream>>>(bufA, mnmx, out, N);
}